// VolumeRenderer_57595511439763
// MI455X (gfx1250) — compile-verified
//
#include <hip/hip_runtime.h>

// Volume rendering (NeRF weight accumulation), MI455X / gfx1250.
// Memory-bound streaming kernel: 161 MB single-pass traffic -> target is HBM
// bandwidth (23.3 TB/s). Wave-per-ray with triple-buffered async global->LDS
// loads (CDNA5 ASYNCcnt path, 2 rays in flight per wave), wave32 shuffle scan
// for transmittance.

#define N_PTS        128
#define FEAT         3
#define FAR_DELTA    1e10f

#define WAVES_PER_BLOCK 8
#define RAYS_PER_WAVE   8
#define DEPTH_BYTES     (N_PTS * 4)            // 512
#define DENS_BYTES      (N_PTS * 4)            // 512
#define FEAT_BYTES      (N_PTS * FEAT * 4)     // 1536
#define BYTES_PER_RAY   (DEPTH_BYTES + DENS_BYTES + FEAT_BYTES)  // 2560
#define NBUF            3                      // 2 rays prefetched ahead

typedef __attribute__((address_space(3))) unsigned char lds_byte_t;

// Workgroup-relative LDS byte offset of a generic pointer to __shared__ data.
__device__ __forceinline__ unsigned lds_off(const void* p) {
    return (unsigned)(unsigned long long)(lds_byte_t*)p;
}

// CDNA5 async copy: 16 bytes per lane, global -> LDS, tracked by ASYNCcnt.
__device__ __forceinline__ void async_ld_b128(unsigned lds_addr, const void* gaddr) {
    asm volatile("global_load_async_to_lds_b128 %0, %1, off"
                 :: "v"(lds_addr), "v"(gaddr)
                 : "memory");
}

__device__ __forceinline__ void wait_async_le10() {   // 2 rays (10 ops) may remain
    asm volatile("s_wait_asynccnt 0xa" ::: "memory");
}
__device__ __forceinline__ void wait_async_le5() {    // 1 ray (5 ops) may remain
    asm volatile("s_wait_asynccnt 0x5" ::: "memory");
}
__device__ __forceinline__ void wait_async_0() {
    asm volatile("s_wait_asynccnt 0x0" ::: "memory");
}
__device__ __forceinline__ void wait_ds_0() {
    asm volatile("s_wait_dscnt 0x0" ::: "memory");
}

// Stage one whole ray (2560 B) into an LDS buffer: 5 fully-coalesced b128
// async loads (each moves a contiguous 512 B wave-segment).
__device__ __forceinline__ void issue_ray(unsigned buf, int lane,
                                          const float* __restrict__ depth,
                                          const float* __restrict__ dens,
                                          const float* __restrict__ feat,
                                          int ray) {
    const char* gd = (const char*)(depth + (size_t)ray * N_PTS);
    const char* gs = (const char*)(dens  + (size_t)ray * N_PTS);
    const char* gf = (const char*)(feat  + (size_t)ray * (N_PTS * FEAT));
    const unsigned lo = (unsigned)lane * 16u;
    async_ld_b128(buf +                                   lo, gd + lo);
    async_ld_b128(buf + DEPTH_BYTES                     + lo, gs + lo);
    async_ld_b128(buf + DEPTH_BYTES + DENS_BYTES        + lo, gf + lo);
    async_ld_b128(buf + DEPTH_BYTES + DENS_BYTES +  512 + lo, gf +  512 + lo);
    async_ld_b128(buf + DEPTH_BYTES + DENS_BYTES + 1024 + lo, gf + 1024 + lo);
}

__global__ __launch_bounds__(WAVES_PER_BLOCK * 32)
void volrend_kernel(const float* __restrict__ depth,
                    const float* __restrict__ dens,
                    const float* __restrict__ feat,
                    float* __restrict__ out,
                    int n_rays) {
    alignas(16) __shared__ unsigned char smem[WAVES_PER_BLOCK * NBUF * BYTES_PER_RAY];
    // Make the LDS region visible to the asm memory clobbers (prevents the
    // compiler from assuming smem is never written).
    asm volatile("" :: "v"((void*)&smem[0]) : "memory");

    const int lane = (int)(threadIdx.x & 31u);
    const int wave = (int)(threadIdx.x >> 5u);
    const int ray0 = ((int)blockIdx.x * WAVES_PER_BLOCK + wave) * RAYS_PER_WAVE;
    if (ray0 >= n_rays) return;

    const unsigned wbase = lds_off(smem) + (unsigned)wave * (NBUF * BYTES_PER_RAY);
    unsigned char* wptr = smem + (size_t)wave * (NBUF * BYTES_PER_RAY);

    // Pipeline prologue: two rays in flight.
    issue_ray(wbase, lane, depth, dens, feat, ray0);
    if (RAYS_PER_WAVE > 1 && ray0 + 1 < n_rays)
        issue_ray(wbase + BYTES_PER_RAY, lane, depth, dens, feat, ray0 + 1);

    int b_cur = 0;  // rotating buffer index 0..2
    for (int i = 0; i < RAYS_PER_WAVE; ++i) {
        const int ray = ray0 + i;
        if (ray >= n_rays) break;

        const bool has1 = (i + 1 < RAYS_PER_WAVE) && (ray + 1 < n_rays);
        const bool has2 = (i + 2 < RAYS_PER_WAVE) && (ray + 2 < n_rays);
        if (has2) {
            int b2 = b_cur + 2; if (b2 >= NBUF) b2 -= NBUF;
            wait_ds_0();   // reads of the buffer we are about to refill are done
            issue_ray(wbase + (unsigned)b2 * BYTES_PER_RAY,
                      lane, depth, dens, feat, ray + 2);
            wait_async_le10();  // ray i landed; rays i+1, i+2 still streaming
        } else if (has1) {
            wait_async_le5();   // ray i landed; ray i+1 still streaming
        } else {
            wait_async_0();
        }

        const unsigned char* bufp = wptr + (size_t)b_cur * BYTES_PER_RAY;
        const float4* sD = (const float4*)bufp;                              // 32 x f4
        const float4* sS = (const float4*)(bufp + DEPTH_BYTES);              // 32 x f4
        const float4* sF = (const float4*)(bufp + DEPTH_BYTES + DENS_BYTES); // 96 x f4

        float4 d4 = sD[lane];          // depth[4l..4l+3]
        float4 s4 = sS[lane];          // density[4l..4l+3]
        float4 f0 = sF[3 * lane + 0];  // {p0.r p0.g p0.b p1.r}
        float4 f1 = sF[3 * lane + 1];  // {p1.g p1.b p2.r p2.g}
        float4 f2 = sF[3 * lane + 2];  // {p2.b p3.r p3.g p3.b}

        // delta_k = depth[k+1]-depth[k]; sentinel 1e10 on the ray's last sample
        float dnext = __shfl_down(d4.x, 1, 32);   // depth[4l+4] (lane 31: unused)
        float t0 = s4.x * (d4.y - d4.x);
        float t1 = s4.y * (d4.z - d4.y);
        float t2 = s4.z * (d4.w - d4.z);
        float t3 = (lane == 31) ? (s4.w * FAR_DELTA) : (s4.w * (dnext - d4.w));

        // wave32 inclusive scan of per-lane tau sums -> exclusive prefix
        float loc = t0 + t1 + t2 + t3;
        float sc = loc;
#pragma unroll
        for (int o = 1; o < 32; o <<= 1) {
            float v = __shfl_up(sc, o, 32);
            if (lane >= o) sc += v;
        }
        float excl = sc - loc;

        // weights: w_k = T_k * (1 - exp(-tau_k)),  T_{k+1} = T_k * exp(-tau_k)
        float e0 = __expf(-t0), e1 = __expf(-t1), e2 = __expf(-t2), e3 = __expf(-t3);
        float T  = __expf(-excl);
        float w0 = T * (1.0f - e0); T *= e0;
        float w1 = T * (1.0f - e1); T *= e1;
        float w2 = T * (1.0f - e2); T *= e2;
        float w3 = T * (1.0f - e3);

        float4 a;
        a.x = w0 * f0.x + w1 * f0.w + w2 * f1.z + w3 * f2.y;   // R
        a.y = w0 * f0.y + w1 * f1.x + w2 * f1.w + w3 * f2.z;   // G
        a.z = w0 * f0.z + w1 * f1.y + w2 * f2.x + w3 * f2.w;   // B
        a.w = w0 * d4.x + w1 * d4.y + w2 * d4.z + w3 * d4.w;   // depth

        // wave32 butterfly reduction of the 4 accumulators
#pragma unroll
        for (int o = 16; o > 0; o >>= 1) {
            a.x += __shfl_xor(a.x, o, 32);
            a.y += __shfl_xor(a.y, o, 32);
            a.z += __shfl_xor(a.z, o, 32);
            a.w += __shfl_xor(a.w, o, 32);
        }
        if (lane == 0) {
            ((float4*)out)[ray] = a;   // [rgb, depth] contiguous, 16B store
        }

        b_cur = (b_cur == NBUF - 1) ? 0 : b_cur + 1;
    }
}

extern "C" void kernel_launch(void* const* d_in, const int* in_sizes, int n_in,
                              void* d_out, int out_size, void* d_ws, size_t ws_size,
                              hipStream_t stream) {
    (void)n_in; (void)d_ws; (void)ws_size; (void)out_size;
    const float* depth = (const float*)d_in[0];  // [n_rays, 128] f32
    const float* dens  = (const float*)d_in[1];  // [n_rays, 128] f32
    const float* feat  = (const float*)d_in[2];  // [n_rays, 128, 3] f32
    float* out = (float*)d_out;                  // [n_rays, 4] f32

    const int n_rays = in_sizes[0] / N_PTS;
    const int rays_per_block = WAVES_PER_BLOCK * RAYS_PER_WAVE;  // 64
    const int blocks = (n_rays + rays_per_block - 1) / rays_per_block;

    volrend_kernel<<<blocks, WAVES_PER_BLOCK * 32, 0, stream>>>(
        depth, dens, feat, out, n_rays);
}